// QuantumMLPHead_65481071410640
// MI455X (gfx1250) — compile-verified
//
#include <hip/hip_runtime.h>

// out[b] = prod_{i=1..13} cos(x[b,i] + shift[i])
// (CNOT-ring conjugation of Z0 == Z1*Z2*...*Z13 on the RY product state)
//
// Feature reduction on the matrix pipe in log2-space:
//   V_WMMA_F32_16X16X4_F32 chained 4x (K = 16, padded), B = all-ones,
//   every D column == row-sum of log2|cos|; result = sign * 2^sum.
// Transcendentals use the raw HW ops (v_cos_f32 / v_log_f32 / v_exp_f32):
// inputs are O(1) so no Payne-Hanek reduction is needed.

typedef __attribute__((ext_vector_type(2))) float v2f;
typedef __attribute__((ext_vector_type(8))) float v8f;

#define NQ 14
#define INV_2PI 0.15915494309189535f   // v_cos_f32 computes cos(2*pi*src)

__global__ __launch_bounds__(256) void
qring_expz_kernel(const float* __restrict__ x,
                  const float* __restrict__ shift,
                  float* __restrict__ out, int batch)
{
    const int tid    = blockIdx.x * blockDim.x + threadIdx.x;
    const int lane   = threadIdx.x & 31;
    const int waveId = tid >> 5;
    const int rowBase = waveId << 4;            // 16 batch rows per wave

    // A-matrix (16x4 f32) layout: lane L<16 -> row L,    K = {0,1} in a.x/a.y
    //                             lane L>=16 -> row L-16, K = {2,3}
    const int r   = lane & 15;
    const int kb  = (lane < 16) ? 0 : 2;
    int grow = rowBase + r;
    grow = (grow < batch - 1) ? grow : (batch - 1);   // clamp (full-wave safety)
    const float* xr = x + grow * NQ;

    v8f acc = {0.f, 0.f, 0.f, 0.f, 0.f, 0.f, 0.f, 0.f};
    v2f bones;
    bones.x = 1.0f;
    bones.y = 1.0f;                              // all-ones B => D = row-sums

    int parity = 0;                              // sign parity of this lane's factors

#pragma unroll
    for (int q = 0; q < 4; ++q) {
        v2f a;
#pragma unroll
        for (int j = 0; j < 2; ++j) {
            const int k  = 4 * q + kb + j;       // feature index 0..15
            // branch-free clamp: every lane issues a valid load (no EXEC games)
            const int kc = (k < 1) ? 1 : ((k > NQ - 1) ? (NQ - 1) : k);
            const float t = xr[kc] + shift[kc];
            const float c = __builtin_amdgcn_cosf(t * INV_2PI);    // v_cos_f32
            const bool act = (k >= 1) && (k <= NQ - 1);            // drop qubit 0 + padding
            parity ^= (act && (c < 0.0f)) ? 1 : 0;
            const float lv = __builtin_amdgcn_logf(__builtin_fabsf(c)); // v_log_f32: log2|c|
            a[j] = act ? lv : 0.0f;              // 0 in log2-space == factor 1
        }
        // D = A x B + C   (f32 16x16x4), chained accumulation over K chunks
        acc = __builtin_amdgcn_wmma_f32_16x16x4_f32(
            /*neg_a=*/false, a, /*neg_b=*/false, bones,
            /*c_mod=*/(short)0, acc, /*reuse_a=*/false, /*reuse_b=*/false);
    }

    // Row r's factors are split across lanes r and r+16: merge sign parity.
    parity ^= __shfl_xor(parity, 16, 32);        // lane L now holds parity of row L%16

    // D layout: VGPR v holds row v (lanes 0-15) / row v+8 (lanes 16-31).
    // Lane L (L%16 < 8) is the unique storer of row (L>=16 ? 8 : 0) + L%16.
    const int vsel  = lane & 15;
    const int v     = vsel & 7;
    const int base8 = (lane < 16) ? 0 : 8;
    const int row16 = base8 + v;
    const int pr    = __shfl(parity, row16, 32); // parity of row16 lives at lane row16

    float sum = acc[v];                          // runtime extract: small cndmask tree
    float res = __builtin_amdgcn_exp2f(sum);     // v_exp_f32: 2^sum == product
    res = pr ? -res : res;

    if (vsel < 8) {
        const int orow = rowBase + row16;
        if (orow < batch) out[orow] = res;
    }
}

extern "C" void kernel_launch(void* const* d_in, const int* in_sizes, int n_in,
                              void* d_out, int out_size, void* d_ws, size_t ws_size,
                              hipStream_t stream)
{
    const float* x     = (const float*)d_in[0];   // [batch, 14] f32
    const float* shift = (const float*)d_in[1];   // [14] f32
    float* out = (float*)d_out;                   // [batch] f32

    const int batch  = in_sizes[0] / NQ;          // 4096
    const int waves  = (batch + 15) / 16;         // 16 rows per wave
    const int thr    = waves * 32;
    const int block  = 256;
    const int blocks = (thr + block - 1) / block; // 32 blocks for batch=4096

    hipLaunchKernelGGL(qring_expz_kernel, dim3(blocks), dim3(block), 0, stream,
                       x, shift, out, batch);
}